// SS2D_69269232550329
// MI455X (gfx1250) — compile-verified
//
#include <hip/hip_runtime.h>
#include <math.h>

typedef __attribute__((ext_vector_type(2))) float v2f;
typedef __attribute__((ext_vector_type(8))) float v8f;

#define D_MODEL 96
#define D_STATE 16
#define D_INNER 192
#define DT_RANK 6
#define NSEQ   128   // sequences per direction (B*H or B*W)
#define SEQL   64    // sequence length
#define NTOK   8192  // tokens per direction
#define XZ_LD  384   // 2*D_INNER
#define XDBL_LD 48   // 38 padded to 48

// Map direction-order row r (= seq*64 + pos) to natural flat token index
// (b*H*W + h*W + w) with B=2, H=W=64.
__device__ __forceinline__ int tok_of_row(int dir, int r) {
    int s = r >> 6;       // sequence
    int l = r & 63;       // position in sequence
    if (dir == 0) return (s << 6) + l;            // h-forward
    if (dir == 1) return (s << 6) + (63 - l);     // h-backward
    int b = s >> 6, w = s & 63;                   // vertical: s = b*64 + w, l = h
    if (dir == 2) return b * 4096 + l * 64 + w;   // v-forward
    return b * 4096 + (63 - l) * 64 + w;          // v-backward
}

// ---------------- Kernel 1: xz[d] = gather(x, dir d) @ in_w[d] ----------------
// One wave per 16x16 tile.  M=8192 (512 tiles), N=384 (24 tiles), K=96.
__global__ void k_inproj(const float* __restrict__ x, const float* __restrict__ in_w,
                         float* __restrict__ xz) {
    int wave = (blockIdx.x * blockDim.x + threadIdx.x) >> 5;
    int lane = threadIdx.x & 31;
    const int TN = XZ_LD / 16, TM = NTOK / 16;
    int d   = wave / (TM * TN);
    int rem = wave % (TM * TN);
    int i = rem / TN, j = rem % TN;
    int half = lane >> 4, mn = lane & 15;

    int t = tok_of_row(d, i * 16 + mn);
    const float* pa = x + (size_t)t * D_MODEL;
    const float* pb = in_w + (size_t)d * D_MODEL * XZ_LD + j * 16 + mn;

    v8f acc = {};
    for (int k0 = 0; k0 < D_MODEL; k0 += 4) {
        int ka = k0 + 2 * half;
        v2f a, b;
        a.x = pa[ka];            a.y = pa[ka + 1];
        b.x = pb[(size_t)ka * XZ_LD];
        b.y = pb[(size_t)(ka + 1) * XZ_LD];
        acc = __builtin_amdgcn_wmma_f32_16x16x4_f32(false, a, false, b,
                                                    (short)0, acc, false, false);
    }
    float* pc = xz + ((size_t)d * NTOK + (size_t)i * 16) * XZ_LD + j * 16 + mn;
    for (int r = 0; r < 8; ++r)
        pc[(size_t)(r + 8 * half) * XZ_LD] = acc[r];
}

// ------------- Kernel 2: depthwise causal conv (K=4) + bias + SiLU, in place ----
// One thread per (dir, seq, channel); register sliding window, safe in-place.
__global__ void k_conv(float* __restrict__ xz, const float* __restrict__ conv_w,
                       const float* __restrict__ conv_b) {
    int idx = blockIdx.x * blockDim.x + threadIdx.x;
    if (idx >= 4 * NSEQ * D_INNER) return;
    int d   = idx / (NSEQ * D_INNER);
    int rem = idx % (NSEQ * D_INNER);
    int s = rem / D_INNER, c = rem % D_INNER;

    const float* w = conv_w + ((size_t)d * D_INNER + c) * 4;
    float w0 = w[0], w1 = w[1], w2 = w[2], w3 = w[3];
    float bias = conv_b[(size_t)d * D_INNER + c];
    float* col = xz + ((size_t)d * NTOK + (size_t)s * SEQL) * XZ_LD + c;

    float x0 = 0.f, x1 = 0.f, x2 = 0.f;
    for (int l = 0; l < SEQL; ++l) {
        float x3 = col[(size_t)l * XZ_LD];
        float v = w0 * x0 + w1 * x1 + w2 * x2 + w3 * x3 + bias;
        v = v / (1.0f + __expf(-v));          // SiLU
        col[(size_t)l * XZ_LD] = v;
        x0 = x1; x1 = x2; x2 = x3;
    }
}

// ---------------- Kernel 3: x_dbl[d] = xi_act[d] @ xproj_w[d] ----------------
// M=8192, K=192, N=38 padded to 48 (zero-filled B cols -> zero outputs).
__global__ void k_xproj(const float* __restrict__ xz, const float* __restrict__ xproj_w,
                        float* __restrict__ xdbl) {
    int wave = (blockIdx.x * blockDim.x + threadIdx.x) >> 5;
    int lane = threadIdx.x & 31;
    const int TN = 3, TM = NTOK / 16;
    int d   = wave / (TM * TN);
    int rem = wave % (TM * TN);
    int i = rem / TN, j = rem % TN;
    int half = lane >> 4, mn = lane & 15;

    const float* pa = xz + ((size_t)d * NTOK + i * 16 + mn) * XZ_LD;  // xi half
    int n = j * 16 + mn;
    const float* pb = xproj_w + (size_t)d * D_INNER * 38;
    bool nok = (n < 38);

    v8f acc = {};
    for (int k0 = 0; k0 < D_INNER; k0 += 4) {
        int ka = k0 + 2 * half;
        v2f a, b;
        a.x = pa[ka];  a.y = pa[ka + 1];
        b.x = nok ? pb[(size_t)ka * 38 + n] : 0.f;
        b.y = nok ? pb[(size_t)(ka + 1) * 38 + n] : 0.f;
        acc = __builtin_amdgcn_wmma_f32_16x16x4_f32(false, a, false, b,
                                                    (short)0, acc, false, false);
    }
    float* pc = xdbl + ((size_t)d * NTOK + (size_t)i * 16) * XDBL_LD + j * 16 + mn;
    for (int r = 0; r < 8; ++r)
        pc[(size_t)(r + 8 * half) * XDBL_LD] = acc[r];
}

// ---------------- Kernel 4: selective scan + gating ----------------
// One block per (dir, seq); thread = channel. h[16] state in registers,
// per-step x_dbl row (dt_low|B|C) staged in LDS and broadcast.
__global__ void __launch_bounds__(D_INNER)
k_scan(const float* __restrict__ xz, const float* __restrict__ xdbl,
       const float* __restrict__ dt_w, const float* __restrict__ dt_b,
       const float* __restrict__ A_log, const float* __restrict__ Dp,
       float* __restrict__ ygate) {
    int d = blockIdx.x / NSEQ;
    int s = blockIdx.x % NSEQ;
    int c = threadIdx.x;

    __shared__ float sdbl[40];

    float Av[D_STATE];
    for (int n = 0; n < D_STATE; ++n)
        Av[n] = -__expf(A_log[(((size_t)d * D_INNER) + c) * D_STATE + n]);
    float dtw[DT_RANK];
    for (int jj = 0; jj < DT_RANK; ++jj)
        dtw[jj] = dt_w[((size_t)d * DT_RANK + jj) * D_INNER + c];
    float dtb = dt_b[(size_t)d * D_INNER + c];
    float Dc  = Dp[(size_t)d * D_INNER + c];

    float h[D_STATE];
    for (int n = 0; n < D_STATE; ++n) h[n] = 0.f;

    const float* xzrow = xz   + ((size_t)d * NTOK + (size_t)s * SEQL) * XZ_LD;
    const float* dbrow = xdbl + ((size_t)d * NTOK + (size_t)s * SEQL) * XDBL_LD;
    float*       yrow  = ygate+ ((size_t)d * NTOK + (size_t)s * SEQL) * D_INNER;

    for (int l = 0; l < SEQL; ++l) {
        __syncthreads();
        if (threadIdx.x < 38) sdbl[threadIdx.x] = dbrow[(size_t)l * XDBL_LD + threadIdx.x];
        __syncthreads();

        float u = xzrow[(size_t)l * XZ_LD + c];
        float z = xzrow[(size_t)l * XZ_LD + D_INNER + c];

        float dtv = dtb;
        for (int jj = 0; jj < DT_RANK; ++jj) dtv += sdbl[jj] * dtw[jj];
        dtv = (dtv > 20.f) ? dtv : log1pf(__expf(dtv));   // softplus

        float y = 0.f;
        for (int n = 0; n < D_STATE; ++n) {
            float dA = __expf(dtv * Av[n]);
            h[n] = dA * h[n] + (dtv * sdbl[6 + n]) * u;
            y += h[n] * sdbl[22 + n];
        }
        y += Dc * u;
        float sz = z / (1.0f + __expf(-z));               // SiLU(z)
        yrow[(size_t)l * D_INNER + c] = y * sz;
    }
}

// -------- Kernel 5: per-dir out-proj GEMM, scatter rows to natural order -------
// yv[d] = ygate[d] @ outp_w[d]  (8192x192 @ 192x96), write out_cat[t][d*96+n].
__global__ void k_outproj(const float* __restrict__ ygate, const float* __restrict__ outp_w,
                          float* __restrict__ ocat) {
    int wave = (blockIdx.x * blockDim.x + threadIdx.x) >> 5;
    int lane = threadIdx.x & 31;
    const int TN = D_MODEL / 16, TM = NTOK / 16;   // 6, 512
    int d   = wave / (TM * TN);
    int rem = wave % (TM * TN);
    int i = rem / TN, j = rem % TN;
    int half = lane >> 4, mn = lane & 15;

    const float* pa = ygate + ((size_t)d * NTOK + i * 16 + mn) * D_INNER;
    const float* pb = outp_w + (size_t)d * D_INNER * D_MODEL + j * 16 + mn;

    v8f acc = {};
    for (int k0 = 0; k0 < D_INNER; k0 += 4) {
        int ka = k0 + 2 * half;
        v2f a, b;
        a.x = pa[ka];  a.y = pa[ka + 1];
        b.x = pb[(size_t)ka * D_MODEL];
        b.y = pb[(size_t)(ka + 1) * D_MODEL];
        acc = __builtin_amdgcn_wmma_f32_16x16x4_f32(false, a, false, b,
                                                    (short)0, acc, false, false);
    }
    for (int r = 0; r < 8; ++r) {
        int m = r + 8 * half;
        int t = tok_of_row(d, i * 16 + m);
        ocat[(size_t)t * XZ_LD + d * D_MODEL + j * 16 + mn] = acc[r];
    }
}

// ---------------- Kernel 6: out = out_cat @ out_w + out_b ----------------
// M=8192, K=384, N=96.
__global__ void k_final(const float* __restrict__ ocat, const float* __restrict__ out_w,
                        const float* __restrict__ out_b, float* __restrict__ out) {
    int wave = (blockIdx.x * blockDim.x + threadIdx.x) >> 5;
    int lane = threadIdx.x & 31;
    const int TN = D_MODEL / 16;                    // 6
    int i = wave / TN, j = wave % TN;
    int half = lane >> 4, mn = lane & 15;

    const float* pa = ocat + (size_t)(i * 16 + mn) * XZ_LD;
    const float* pb = out_w + j * 16 + mn;

    v8f acc = {};
    for (int k0 = 0; k0 < XZ_LD; k0 += 4) {
        int ka = k0 + 2 * half;
        v2f a, b;
        a.x = pa[ka];  a.y = pa[ka + 1];
        b.x = pb[(size_t)ka * D_MODEL];
        b.y = pb[(size_t)(ka + 1) * D_MODEL];
        acc = __builtin_amdgcn_wmma_f32_16x16x4_f32(false, a, false, b,
                                                    (short)0, acc, false, false);
    }
    float bias = out_b[j * 16 + mn];
    for (int r = 0; r < 8; ++r)
        out[(size_t)(i * 16 + r + 8 * half) * D_MODEL + j * 16 + mn] = acc[r] + bias;
}

extern "C" void kernel_launch(void* const* d_in, const int* in_sizes, int n_in,
                              void* d_out, int out_size, void* d_ws, size_t ws_size,
                              hipStream_t stream) {
    const float* x       = (const float*)d_in[0];
    const float* in_w    = (const float*)d_in[1];
    const float* conv_w  = (const float*)d_in[2];
    const float* conv_b  = (const float*)d_in[3];
    const float* xproj_w = (const float*)d_in[4];
    const float* dt_w    = (const float*)d_in[5];
    const float* dt_b    = (const float*)d_in[6];
    const float* A_log   = (const float*)d_in[7];
    const float* Dp      = (const float*)d_in[8];
    const float* outp_w  = (const float*)d_in[9];
    const float* out_w   = (const float*)d_in[10];
    const float* out_b   = (const float*)d_in[11];
    float* out = (float*)d_out;

    // Workspace layout (floats). Every region is fully written before read.
    float* ws    = (float*)d_ws;
    float* xz    = ws;                                   // 4*8192*384 = 12,582,912
    float* xdbl  = xz   + (size_t)4 * NTOK * XZ_LD;      // 4*8192*48  =  1,572,864
    float* ygate = xdbl + (size_t)4 * NTOK * XDBL_LD;    // 4*8192*192 =  6,291,456
    float* ocat  = ygate+ (size_t)4 * NTOK * D_INNER;    // 8192*384   =  3,145,728
    (void)ws_size; (void)in_sizes; (void)n_in; (void)out_size;

    // K1: in-projection.  4*512*24 waves, 8 waves/block.
    k_inproj<<<(4 * 512 * 24) / 8, 256, 0, stream>>>(x, in_w, xz);
    // K2: depthwise conv + SiLU (in place on xi half of xz).
    k_conv<<<(4 * NSEQ * D_INNER + 255) / 256, 256, 0, stream>>>(xz, conv_w, conv_b);
    // K3: x projection -> (dt_low | B | C).
    k_xproj<<<(4 * 512 * 3) / 8, 256, 0, stream>>>(xz, xproj_w, xdbl);
    // K4: selective scan + gate.  One block per (dir, seq).
    k_scan<<<4 * NSEQ, D_INNER, 0, stream>>>(xz, xdbl, dt_w, dt_b, A_log, Dp, ygate);
    // K5: per-direction out-proj with scatter to natural token order.
    k_outproj<<<(4 * 512 * 6) / 8, 256, 0, stream>>>(ygate, outp_w, ocat);
    // K6: final merge GEMM + bias.
    k_final<<<(512 * 6) / 8, 256, 0, stream>>>(ocat, out_w, out_b, out);
}